// FermiSets_41669772706327
// MI455X (gfx1250) — compile-verified
//
#include <hip/hip_runtime.h>
#include <hip/hip_bf16.h>
#include <math.h>

typedef __attribute__((ext_vector_type(2))) float v2f;
typedef __attribute__((ext_vector_type(8))) float v8f;

#define NPART 128
#define HID   8

__device__ __forceinline__ float gelu_tanh(float v) {
    // Exact tanh-GELU identity: 0.5*v*(1+tanh(u)) == v * sigmoid(2u),
    // u = sqrt(2/pi)*(v + 0.044715 v^3).  Branch-free via hw exp2 + rcp.
    const float k = 2.0f * 0.7978845608028654f * 1.4426950408889634f; // 2*c0*log2(e)
    const float c1 = 0.044715f;
    float u = fmaf(c1 * v * v, v, v);                 // v + c1*v^3
    float e = __builtin_amdgcn_exp2f(-k * u);         // exp(-2*c0*u)
    return v * __builtin_amdgcn_rcpf(1.0f + e);       // v * sigmoid
}

__global__ __launch_bounds__(128) void fermi_sets_kernel(
    const float* __restrict__ x,
    const float* __restrict__ phi_w, const float* __restrict__ phi_b,
    const float* __restrict__ rho_w, const float* __restrict__ rho_b,
    const float* __restrict__ psi1_w, const float* __restrict__ psi1_b,
    const float* __restrict__ psi2_w, const float* __restrict__ psi2_b,
    float* __restrict__ out)
{
    __shared__ float2 zc[NPART];          // particle positions (complex)
    __shared__ float  wl[228];            // packed weights
    __shared__ float  Spart[4][HID];      // per-wave deepsets partial sums
    __shared__ float2 nupart[4];          // per-wave complex product partials

    const int t    = threadIdx.x;
    const int lane = t & 31;
    const int w    = t >> 5;
    const int b    = blockIdx.x;

    // ---------------- Phase 0: stage inputs in LDS ----------------
    const float2* xg = (const float2*)x + (size_t)b * NPART;
    float2 zt = xg[t];
    zc[t] = zt;
    if (t < 16)  wl[t]       = phi_w[t];                         // [0..15]   phi_w (2x8)
    if (t < 8)   { wl[16+t]  = phi_b[t];  wl[88+t] = rho_b[t]; } // [16..23], [88..95]
    if (t < 64)  wl[24+t]    = rho_w[t];                         // [24..87]  rho_w (8x8)
    if (t < 100) wl[96+t]    = psi1_w[t];                        // [96..195] psi1_w (10x10)
    if (t < 10)  wl[196+t]   = psi1_b[t];                        // [196..205]
    if (t < 20)  wl[206+t]   = psi2_w[t];                        // [206..225] psi2_w (10x2)
    if (t < 2)   wl[226+t]   = psi2_b[t];                        // [226..227]
    __syncthreads();

    // ---------------- Phase B: phi matmul via WMMA f32 16x16x4 ----------------
    // A[16x4]: lanes 0-15 hold K=0,1 (x,y); lanes 16-31 hold K=2,3 = 0 (pad)
    // B[4x16]: phi_w padded; cols >= 8 zero.  All LDS reads unconditional; zero
    // padding applied with selects (no exec-mask branches).
    const int  n      = lane & 15;
    const int  h      = n & 7;
    const bool lo16   = (lane < 16);
    const bool bvalid = lo16 && (n < HID);
    const float bw0 = wl[h];          // phi_w[0][h]
    const float bw1 = wl[8 + h];      // phi_w[1][h]
    v2f bm;
    bm.x = bvalid ? bw0 : 0.f;
    bm.y = bvalid ? bw1 : 0.f;
    const float bias = wl[16 + h];

    float accS = 0.f;
    #pragma unroll
    for (int tile = 0; tile < 2; ++tile) {
        const int rowbase = w * 32 + tile * 16;   // 4 waves x 2 tiles x 16 rows = 128
        const float2 p = zc[rowbase + n];         // rowbase+n <= 127: always in range
        v2f a;
        a.x = lo16 ? p.x : 0.f;
        a.y = lo16 ? p.y : 0.f;
        v8f c = {};
        c = __builtin_amdgcn_wmma_f32_16x16x4_f32(false, a, false, bm,
                                                  (short)0, c, false, false);
        float g = 0.f;
        #pragma unroll
        for (int v = 0; v < 8; ++v) g += gelu_tanh(c[v] + bias);
        accS += g;
    }
    accS += __shfl_xor(accS, 16, 32);   // combine both row halves of each column
    if (lane < HID) Spart[w][lane] = accS;

    // ---------------- Phase A: pairwise product (balanced circular mapping) ----
    // thread t handles unordered pairs {t, (t+k) mod 128}, k=1..63 (+k=64 for t<64).
    // 4 independent complex accumulators break the FMA dependency chain.
    float a0x = 1.f, a0y = 0.f, a1x = 1.f, a1y = 0.f;
    float a2x = 1.f, a2y = 0.f, a3x = 1.f, a3y = 0.f;

    auto pstep = [&](int u, float& px, float& py) {
        const float2 zu = zc[u];
        const float dx = zt.x - zu.x;
        const float dy = zt.y - zu.y;
        const float d2 = fmaf(dx, dx, fmaf(dy, dy, 1.0f));
        const float inv = __builtin_amdgcn_rsqf(d2);
        const float rx = dx * inv, ry = dy * inv;
        const float nx = fmaf(px, rx, -py * ry);
        const float ny = fmaf(px, ry,  py * rx);
        px = nx; py = ny;
    };

    for (int k = 1; k <= 57; k += 4) {
        pstep((t + k)     & (NPART - 1), a0x, a0y);
        pstep((t + k + 1) & (NPART - 1), a1x, a1y);
        pstep((t + k + 2) & (NPART - 1), a2x, a2y);
        pstep((t + k + 3) & (NPART - 1), a3x, a3y);
    }
    pstep((t + 61) & (NPART - 1), a0x, a0y);
    pstep((t + 62) & (NPART - 1), a1x, a1y);
    pstep((t + 63) & (NPART - 1), a2x, a2y);
    if (t < 64) pstep(t + 64, a3x, a3y);   // whole waves 0-1: no intra-wave divergence

    // combine 4 accumulators: (a0*a1)*(a2*a3)
    float e0x = a0x * a1x - a0y * a1y;
    float e0y = fmaf(a0x, a1y, a0y * a1x);
    float e1x = a2x * a3x - a2y * a3y;
    float e1y = fmaf(a2x, a3y, a2y * a3x);
    float axr = e0x * e1x - e0y * e1y;
    float ayr = fmaf(e0x, e1y, e0y * e1x);

    // closed-form sign parity: #flips = |{k in 1..63 : (t+k) mod 128 > t}| (+1 if t<64)
    int par = ((t <= 64) ? 63 : (127 - t)) + ((t < 64) ? 1 : 0);
    if (par & 1) { axr = -axr; ayr = -ayr; }

    // wave32 complex-product butterfly reduction
    #pragma unroll
    for (int off = 16; off >= 1; off >>= 1) {
        const float bx = __shfl_xor(axr, off, 32);
        const float by = __shfl_xor(ayr, off, 32);
        const float nx = axr * bx - ayr * by;
        const float ny = fmaf(axr, by, ayr * bx);
        axr = nx; ayr = ny;
    }
    if (lane == 0) nupart[w] = make_float2(axr, ayr);
    __syncthreads();

    // ---------------- Phase C: tail MLP (tiny; thread 0) ----------------
    if (t == 0) {
        float nre = nupart[0].x, nim = nupart[0].y;
        #pragma unroll
        for (int i = 1; i < 4; ++i) {
            const float br = nupart[i].x, bi = nupart[i].y;
            const float nr = nre * br - nim * bi;
            const float ni = fmaf(nre, bi, nim * br);
            nre = nr; nim = ni;
        }
        float S[HID];
        #pragma unroll
        for (int j = 0; j < HID; ++j)
            S[j] = Spart[0][j] + Spart[1][j] + Spart[2][j] + Spart[3][j];

        // rho layer: y2 = gelu(S @ rho_w + rho_b)
        float y2[HID];
        #pragma unroll
        for (int j = 0; j < HID; ++j) {
            float acc = wl[88 + j];
            #pragma unroll
            for (int k = 0; k < HID; ++k) acc = fmaf(S[k], wl[24 + k * HID + j], acc);
            y2[j] = gelu_tanh(acc);
        }

        float res[2][2];
        #pragma unroll
        for (int s = 0; s < 2; ++s) {
            const float sgn = s ? -1.f : 1.f;
            float feat[10];
            #pragma unroll
            for (int j = 0; j < HID; ++j) feat[j] = y2[j];
            feat[8] = sgn * nre;
            feat[9] = sgn * nim;
            float p[10];
            #pragma unroll
            for (int j = 0; j < 10; ++j) {
                float acc = wl[196 + j];
                #pragma unroll
                for (int k = 0; k < 10; ++k) acc = fmaf(feat[k], wl[96 + k * 10 + j], acc);
                p[j] = gelu_tanh(acc);
            }
            float o0 = wl[226], o1 = wl[227];
            #pragma unroll
            for (int k = 0; k < 10; ++k) {
                o0 = fmaf(p[k], wl[206 + 2 * k],     o0);
                o1 = fmaf(p[k], wl[206 + 2 * k + 1], o1);
            }
            res[s][0] = o0; res[s][1] = o1;
        }
        out[2 * b]     = 0.5f * (res[0][0] - res[1][0]);
        out[2 * b + 1] = 0.5f * (res[0][1] - res[1][1]);
    }
}

extern "C" void kernel_launch(void* const* d_in, const int* in_sizes, int n_in,
                              void* d_out, int out_size, void* d_ws, size_t ws_size,
                              hipStream_t stream) {
    (void)in_sizes; (void)n_in; (void)d_ws; (void)ws_size; (void)out_size;
    const float* xin    = (const float*)d_in[0];
    const float* phi_w  = (const float*)d_in[1];
    const float* phi_b  = (const float*)d_in[2];
    const float* rho_w  = (const float*)d_in[3];
    const float* rho_b  = (const float*)d_in[4];
    const float* psi1_w = (const float*)d_in[5];
    const float* psi1_b = (const float*)d_in[6];
    const float* psi2_w = (const float*)d_in[7];
    const float* psi2_b = (const float*)d_in[8];
    float* out = (float*)d_out;

    const int BATCH = 2048;
    fermi_sets_kernel<<<BATCH, 128, 0, stream>>>(
        xin, phi_w, phi_b, rho_w, rho_b, psi1_w, psi1_b, psi2_w, psi2_b, out);
}